// MultimodalFusionGAT_43061342110394
// MI455X (gfx1250) — compile-verified
//
#include <hip/hip_runtime.h>
#include <hip/hip_bf16.h>
#include <math.h>

typedef float v2f __attribute__((ext_vector_type(2)));
typedef float v8f __attribute__((ext_vector_type(8)));

#define NEG_SLOPE 0.2f
#define SEG_EPS   1e-16f
#define MAXC      16

// ---------- helpers: order-preserving float<->int encoding for atomic max ----------
__device__ __forceinline__ int f32_to_ordered(float f) {
    int b = __float_as_int(f);
    return b >= 0 ? b : (b ^ 0x7FFFFFFF);
}
__device__ __forceinline__ float ordered_to_f32(int k) {
    int b = k >= 0 ? k : (k ^ 0x7FFFFFFF);
    return __int_as_float(b);
}

// ---------- 0) build int32 src/dst with self loops appended ----------
__global__ void k_build_edges(const long long* __restrict__ edge64,
                              int* __restrict__ src, int* __restrict__ dst,
                              int E, int E2) {
    int i = blockIdx.x * blockDim.x + threadIdx.x;
    if (i >= E2) return;
    if (i < E) {
        src[i] = (int)edge64[i];
        dst[i] = (int)edge64[(long long)E + i];
    } else {
        int n = i - E;           // self loop
        src[i] = n;
        dst[i] = n;
    }
}

// ---------- 1) init logits = fc_b ----------
__global__ void k_init_logits(float* __restrict__ logits, const float* __restrict__ fcb,
                              int N, int C) {
    int i = blockIdx.x * blockDim.x + threadIdx.x;
    if (i >= N * C) return;
    logits[i] = fcb[i % C];
}

// ---------- 2) WMMA fp32 GEMM: h[N,H] = x[N,K] @ W[K,H] ----------
// block = 256 threads = 8 waves. ntiles = H/16 tiles across columns,
// waves_m = 8/ntiles tiles of 16 rows per block.
__global__ void k_gemm_h(const float* __restrict__ x, const float* __restrict__ W,
                         float* __restrict__ h, int N, int K, int H) {
    const int ntiles  = H >> 4;            // tiles in N-dim (4 for H=64)
    const int waves_m = 8 / ntiles;        // row tiles per block (2 for H=64)
    int wave  = threadIdx.x >> 5;
    int lane  = threadIdx.x & 31;
    int l16   = lane & 15;
    int khalf = lane >> 4;                 // 0: K=0,1  1: K=2,3

    int mbase = blockIdx.x * (16 * waves_m) + (wave / ntiles) * 16;
    int nbase = (wave % ntiles) * 16;

    int arow = mbase + l16;
    if (arow >= N) arow = N - 1;           // clamp: keep EXEC all-ones for WMMA
    int col  = nbase + l16;

    const float* xrow = x + (size_t)arow * K + 2 * khalf;
    const float* wp   = W + (size_t)(2 * khalf) * H + col;

    v8f acc = {0.f, 0.f, 0.f, 0.f, 0.f, 0.f, 0.f, 0.f};

    for (int k = 0; k < K; k += 4) {
        v2f a = *(const v2f*)(xrow + k);                 // A[m][k+2kh], A[m][k+2kh+1]
        v2f b;
        b.x = wp[(size_t)k * H];                         // B[k+2kh][n]
        b.y = wp[(size_t)(k + 1) * H];                   // B[k+2kh+1][n]
        acc = __builtin_amdgcn_wmma_f32_16x16x4_f32(
            false, a, false, b, (short)0, acc, false, false);
    }

    // C/D layout: VGPR j -> row mbase + j + 8*khalf, col = nbase + l16
    int r0 = mbase + 8 * khalf;
#pragma unroll
    for (int j = 0; j < 8; ++j) {
        int r = r0 + j;
        if (r < N) h[(size_t)r * H + col] = acc[j];
    }
}

// ---------- 3) alphas + per-modality init (m, denom, out) ----------
__global__ void k_alpha_init(const float* __restrict__ h,
                             const float* __restrict__ att_src,
                             const float* __restrict__ att_dst,
                             float* __restrict__ alpha_s, float* __restrict__ alpha_d,
                             float* __restrict__ denom, int* __restrict__ m_ord,
                             float* __restrict__ out, int N, int H) {
    int i = blockIdx.x * blockDim.x + threadIdx.x;
    if (i >= N) return;
    const float* hr = h + (size_t)i * H;
    float s = 0.f, d = 0.f;
    for (int c = 0; c < H; ++c) {
        float v = hr[c];
        s += v * att_src[c];
        d += v * att_dst[c];
    }
    alpha_s[i] = s;
    alpha_d[i] = d;
    denom[i] = 0.f;
    m_ord[i] = f32_to_ordered(-INFINITY);
    float* orow = out + (size_t)i * H;
    for (int c = 0; c < H; ++c) orow[c] = 0.f;
}

// ---------- 4) edge pass 1: e = lrelu(as[src]+ad[dst]); segment max ----------
__global__ void k_edge_max(const int* __restrict__ src, const int* __restrict__ dst,
                           const float* __restrict__ as_, const float* __restrict__ ad_,
                           float* __restrict__ ebuf, int* __restrict__ m_ord, int E2) {
    int i = blockIdx.x * blockDim.x + threadIdx.x;
    if (i >= E2) return;
    int s = src[i], d = dst[i];
    float e = as_[s] + ad_[d];
    e = e > 0.f ? e : NEG_SLOPE * e;
    ebuf[i] = e;
    atomicMax(&m_ord[d], f32_to_ordered(e));
}

// ---------- 5) edge pass 2: ex = exp(e - m[dst]); segment sum ----------
__global__ void k_edge_exp(const int* __restrict__ dst, float* __restrict__ ebuf,
                           const int* __restrict__ m_ord, float* __restrict__ denom, int E2) {
    int i = blockIdx.x * blockDim.x + threadIdx.x;
    if (i >= E2) return;
    int d = dst[i];
    float m = ordered_to_f32(m_ord[d]);
    float ex = expf(ebuf[i] - m);
    ebuf[i] = ex;
    atomicAdd(&denom[d], ex);
}

// ---------- 6) edge pass 3: out[dst] += attn * h[src]  (one wave per edge) ----------
__global__ void k_edge_scatter(const int* __restrict__ src, const int* __restrict__ dst,
                               const float* __restrict__ ebuf, const float* __restrict__ denom,
                               const float* __restrict__ h, float* __restrict__ out,
                               int E2, int H) {
    int gid  = blockIdx.x * blockDim.x + threadIdx.x;
    int edge = gid >> 5;
    int lane = gid & 31;
    if (edge >= E2) return;
    int s = src[edge], d = dst[edge];
    float attn = ebuf[edge] / (denom[d] + SEG_EPS);
    const float* hs = h + (size_t)s * H;
    float* od = out + (size_t)d * H;
    for (int c = 2 * lane; c < H; c += 64) {
        v2f v = *(const v2f*)(hs + c);
        atomicAdd(&od[c],     attn * v.x);
        atomicAdd(&od[c + 1], attn * v.y);
    }
}

// ---------- 7) logits += (out + b_gat) @ fc_W[mod*H : (mod+1)*H, :] ----------
__global__ void k_fc_accum(const float* __restrict__ out, const float* __restrict__ b_gat,
                           const float* __restrict__ fcW, float* __restrict__ logits,
                           int N, int H, int C, int mod) {
    int i = blockIdx.x * blockDim.x + threadIdx.x;
    if (i >= N) return;
    const float* orow = out + (size_t)i * H;
    float acc[MAXC];
    for (int c = 0; c < C; ++c) acc[c] = 0.f;
    for (int k = 0; k < H; ++k) {
        float v = orow[k] + b_gat[k];
        const float* wr = fcW + (size_t)(mod * H + k) * C;
        for (int c = 0; c < C; ++c) acc[c] += v * wr[c];
    }
    float* lr = logits + (size_t)i * C;
    for (int c = 0; c < C; ++c) lr[c] += acc[c];
}

// ---------- 8) in-place log_softmax over C ----------
__global__ void k_log_softmax(float* __restrict__ logits, int N, int C) {
    int i = blockIdx.x * blockDim.x + threadIdx.x;
    if (i >= N) return;
    float* lr = logits + (size_t)i * C;
    float m = lr[0];
    for (int c = 1; c < C; ++c) m = fmaxf(m, lr[c]);
    float sum = 0.f;
    for (int c = 0; c < C; ++c) sum += expf(lr[c] - m);
    float lse = m + logf(sum);
    for (int c = 0; c < C; ++c) lr[c] = lr[c] - lse;
}

extern "C" void kernel_launch(void* const* d_in, const int* in_sizes, int n_in,
                              void* d_out, int out_size, void* d_ws, size_t ws_size,
                              hipStream_t stream) {
    const float* x[3] = {(const float*)d_in[0], (const float*)d_in[1], (const float*)d_in[2]};
    const long long* edge64 = (const long long*)d_in[3];
    const float* W       = (const float*)d_in[4];
    const float* att_src = (const float*)d_in[5];
    const float* att_dst = (const float*)d_in[6];
    const float* b_gat   = (const float*)d_in[7];
    const float* fcW     = (const float*)d_in[8];
    const float* fcb     = (const float*)d_in[9];
    float* logits = (float*)d_out;

    const int H  = in_sizes[5];                 // 64
    const int K  = in_sizes[4] / H;             // 256
    const int N  = in_sizes[0] / K;             // 50000
    const int E  = in_sizes[3] / 2;             // 1.6M
    const int C  = in_sizes[9];                 // 10
    const int E2 = E + N;                       // + self loops

    // ---- workspace carve ----
    char* ws = (char*)d_ws;
    size_t off = 0;
    auto carve = [&](size_t bytes) -> void* {
        void* p = ws + off;
        off += (bytes + 255) & ~(size_t)255;
        return p;
    };
    int*   src32   = (int*)  carve((size_t)E2 * 4);
    int*   dst32   = (int*)  carve((size_t)E2 * 4);
    float* hbuf    = (float*)carve((size_t)N * H * 4);
    float* outbuf  = (float*)carve((size_t)N * H * 4);
    float* alpha_s = (float*)carve((size_t)N * 4);
    float* alpha_d = (float*)carve((size_t)N * 4);
    float* denom   = (float*)carve((size_t)N * 4);
    int*   m_ord   = (int*)  carve((size_t)N * 4);
    float* ebuf    = (float*)carve((size_t)E2 * 4);
    (void)ws_size;

    const int T = 256;
    dim3 blkE((E2 + T - 1) / T);
    dim3 blkN((N + T - 1) / T);
    dim3 blkNC((N * C + T - 1) / T);
    long long scatter_threads = (long long)E2 * 32;
    dim3 blkScat((unsigned)((scatter_threads + T - 1) / T));
    const int waves_m = 8 / (H >> 4);           // 2 for H=64
    dim3 blkGemm((N + 16 * waves_m - 1) / (16 * waves_m));

    k_build_edges<<<blkE, T, 0, stream>>>(edge64, src32, dst32, E, E2);
    k_init_logits<<<blkNC, T, 0, stream>>>(logits, fcb, N, C);

    for (int mod = 0; mod < 3; ++mod) {
        k_gemm_h<<<blkGemm, T, 0, stream>>>(x[mod], W, hbuf, N, K, H);
        k_alpha_init<<<blkN, T, 0, stream>>>(hbuf, att_src, att_dst,
                                             alpha_s, alpha_d, denom, m_ord, outbuf, N, H);
        k_edge_max<<<blkE, T, 0, stream>>>(src32, dst32, alpha_s, alpha_d, ebuf, m_ord, E2);
        k_edge_exp<<<blkE, T, 0, stream>>>(dst32, ebuf, m_ord, denom, E2);
        k_edge_scatter<<<blkScat, T, 0, stream>>>(src32, dst32, ebuf, denom, hbuf, outbuf, E2, H);
        k_fc_accum<<<blkN, T, 0, stream>>>(outbuf, b_gat, fcW, logits, N, H, C, mod);
    }

    k_log_softmax<<<blkN, T, 0, stream>>>(logits, N, C);
}